// Block_35012573397335
// MI455X (gfx1250) — compile-verified
//
#include <hip/hip_runtime.h>

// ---------------------------------------------------------------------------
// GNN encoder (3 × [SAGEConv-gcn + GraphConv] + FC) for MI455X / gfx1250.
// Aggregations: fp32 global atomics (L2-resident, 192MB); GEMMs: fp32 WMMA
// v_wmma_f32_16x16x4_f32 with fused normalization + bias + ReLU epilogues.
// ---------------------------------------------------------------------------

typedef __attribute__((ext_vector_type(2))) float v2f;
typedef __attribute__((ext_vector_type(8))) float v8f;

#define NN 50000          // nodes (fixed by the reference)
#define ROWTILES 3125     // 50000 / 16

// -------------------------------- degrees ----------------------------------
__global__ __launch_bounds__(256)
void deg_kernel(const int* __restrict__ src, const int* __restrict__ dst,
                float* __restrict__ in_deg, float* __restrict__ out_deg, int E) {
    int e = blockIdx.x * 256 + threadIdx.x;
    if (e >= E) return;
    unsafeAtomicAdd(&out_deg[src[e]], 1.0f);
    unsafeAtomicAdd(&in_deg[dst[e]], 1.0f);
}

__global__ __launch_bounds__(256)
void scal_kernel(const float* __restrict__ in_deg, const float* __restrict__ out_deg,
                 float* __restrict__ inv_src, float* __restrict__ inv_dst,
                 float* __restrict__ rden, int n) {
    int i = blockIdx.x * 256 + threadIdx.x;
    if (i >= n) return;
    float od = out_deg[i];
    float id = in_deg[i];
    inv_src[i] = rsqrtf(od > 0.0f ? od : 1.0f);
    inv_dst[i] = rsqrtf(id > 0.0f ? id : 1.0f);
    rden[i]    = 1.0f / (id + 1.0f);
}

// ------------------------------ edge scatter -------------------------------
// agg[dst[e], :] += h[src[e], :] * (SCALED ? scale[src[e]] : 1)
// One thread per (edge, float4 chunk); DIM compile-time -> cheap div/mod.
template<int DIM, bool SCALED>
__global__ __launch_bounds__(256)
void scatter_kernel(const float* __restrict__ h, const float* __restrict__ scale,
                    const int* __restrict__ src, const int* __restrict__ dst,
                    float* __restrict__ agg, int nEdges) {
    constexpr unsigned DIM4 = DIM / 4;
    unsigned gid = blockIdx.x * 256u + threadIdx.x;
    unsigned e = gid / DIM4;
    if (e >= (unsigned)nEdges) return;
    unsigned c = (gid - e * DIM4) * 4u;
    int s = src[e];
    int d = dst[e];
    float4 v = *reinterpret_cast<const float4*>(h + (size_t)s * DIM + c);
    if (SCALED) {
        float sc = scale[s];
        v.x *= sc; v.y *= sc; v.z *= sc; v.w *= sc;
    }
    float* a = agg + (size_t)d * DIM + c;
    unsafeAtomicAdd(a + 0, v.x);
    unsafeAtomicAdd(a + 1, v.y);
    unsafeAtomicAdd(a + 2, v.z);
    unsafeAtomicAdd(a + 3, v.w);
}

// ------------------------------ WMMA GEMM ----------------------------------
// out[n, dout] = epilogue( A[n, din] @ W[dout, din]^T + bias )
// MODE 0 (SAGE): A = (A1 + A2) * scale[row]   (agg + h) / (in_deg+1)
// MODE 1 (GC):   A = A1 * scale[row]          agg * in_deg^-1/2
// MODE 2 (FC):   A = A1
// One wave per 16x16 C tile; K-loop of v_wmma_f32_16x16x4_f32.
// Fragment layouts per CDNA5 ISA 7.12.2 (32-bit A 16x4 / B 4x16 / C 16x16).
template<int MODE>
__global__ __launch_bounds__(256)
void gemm_kernel(const float* __restrict__ A1, const float* __restrict__ A2,
                 const float* __restrict__ scale, const float* __restrict__ W,
                 const float* __restrict__ bias, float* __restrict__ out,
                 int din, int dout, int relu) {
    const int lane  = threadIdx.x & 31;
    const int wave  = threadIdx.x >> 5;
    const int tileN = blockIdx.x;                 // column tile
    const int tileM = blockIdx.y * 8 + wave;      // row tile (wave-uniform)
    if (tileM >= ROWTILES) return;                // whole wave exits: EXEC stays all-1s

    const int mrow = tileM * 16 + (lane & 15);    // A row fed by this lane
    const int col  = tileN * 16 + (lane & 15);    // B column / C column
    const int koff = (lane >> 4) << 1;            // lanes 0-15: K+{0,1}; 16-31: K+{2,3}

    const size_t arow = (size_t)mrow * din;
    const size_t wrow = (size_t)col  * din;

    float s = 1.0f;
    if (MODE != 2) s = scale[mrow];

    v8f acc = {};
    for (int k0 = 0; k0 < din; k0 += 4) {
        const int k = k0 + koff;                  // k even -> aligned float2
        float2 b2 = *reinterpret_cast<const float2*>(W + wrow + k);
        v2f a, b;
        b.x = b2.x; b.y = b2.y;
        if (MODE == 0) {
            float2 g  = *reinterpret_cast<const float2*>(A1 + arow + k);
            float2 h2 = *reinterpret_cast<const float2*>(A2 + arow + k);
            a.x = (g.x + h2.x) * s;
            a.y = (g.y + h2.y) * s;
        } else if (MODE == 1) {
            float2 g = *reinterpret_cast<const float2*>(A1 + arow + k);
            a.x = g.x * s;
            a.y = g.y * s;
        } else {
            float2 g = *reinterpret_cast<const float2*>(A1 + arow + k);
            a.x = g.x;
            a.y = g.y;
        }
        acc = __builtin_amdgcn_wmma_f32_16x16x4_f32(
                  false, a, false, b, (short)0, acc, false, false);
    }

    const float bv = bias[col];
    const int rbase = tileM * 16 + ((lane >> 4) << 3);   // C rows r + 8*(lane>=16)
#pragma unroll
    for (int r = 0; r < 8; ++r) {
        float v = acc[r] + bv;
        if (relu) v = fmaxf(v, 0.0f);
        out[(size_t)(rbase + r) * dout + col] = v;
    }
}

// ------------------------------- launcher ----------------------------------
extern "C" void kernel_launch(void* const* d_in, const int* in_sizes, int n_in,
                              void* d_out, int out_size, void* d_ws, size_t ws_size,
                              hipStream_t stream) {
    const int N = NN;
    const int E = in_sizes[1];

    const float* features = (const float*)d_in[0];
    const int*   src      = (const int*)d_in[1];
    const int*   dst      = (const int*)d_in[2];
    const float* sw0 = (const float*)d_in[3],  *sb0 = (const float*)d_in[4];
    const float* gw0 = (const float*)d_in[5],  *gb0 = (const float*)d_in[6];
    const float* sw1 = (const float*)d_in[7],  *sb1 = (const float*)d_in[8];
    const float* gw1 = (const float*)d_in[9],  *gb1 = (const float*)d_in[10];
    const float* sw2 = (const float*)d_in[11], *sb2 = (const float*)d_in[12];
    const float* gw2 = (const float*)d_in[13], *gb2 = (const float*)d_in[14];
    const float* fcw = (const float*)d_in[15], *fcb = (const float*)d_in[16];

    // workspace layout (floats); total = (5 + 512 + 384 + 384) * N ~= 257 MB
    float* ws      = (float*)d_ws;
    float* in_deg  = ws;
    float* out_deg = ws + (size_t)N;
    float* inv_src = ws + 2 * (size_t)N;
    float* inv_dst = ws + 3 * (size_t)N;
    float* rden    = ws + 4 * (size_t)N;
    float* agg     = ws + 5 * (size_t)N;            // N x 512
    float* hA      = agg + (size_t)N * 512;         // N x 384
    float* hB      = hA  + (size_t)N * 384;         // N x 384

    const dim3 blk(256);
    const unsigned gy = (ROWTILES + 7) / 8;         // 391 row-tile groups (8 waves/block)

    // degrees + scalers
    hipMemsetAsync(in_deg, 0, 2 * (size_t)N * sizeof(float), stream);
    deg_kernel<<<(E + 255) / 256, blk, 0, stream>>>(src, dst, in_deg, out_deg, E);
    scal_kernel<<<(N + 255) / 256, blk, 0, stream>>>(in_deg, out_deg, inv_src, inv_dst, rden, N);

    // ---- stage 0: SAGE 512 -> 384 ----
    hipMemsetAsync(agg, 0, (size_t)N * 512 * sizeof(float), stream);
    scatter_kernel<512, false><<<(unsigned)(((long)E * 128 + 255) / 256), blk, 0, stream>>>(
        features, nullptr, src, dst, agg, E);
    gemm_kernel<0><<<dim3(384 / 16, gy), blk, 0, stream>>>(agg, features, rden, sw0, sb0, hA, 512, 384, 1);

    // ---- stage 0: GraphConv 384 -> 384 ----
    hipMemsetAsync(agg, 0, (size_t)N * 384 * sizeof(float), stream);
    scatter_kernel<384, true><<<(unsigned)(((long)E * 96 + 255) / 256), blk, 0, stream>>>(
        hA, inv_src, src, dst, agg, E);
    gemm_kernel<1><<<dim3(384 / 16, gy), blk, 0, stream>>>(agg, nullptr, inv_dst, gw0, gb0, hB, 384, 384, 1);

    // ---- stage 1: SAGE 384 -> 256 ----
    hipMemsetAsync(agg, 0, (size_t)N * 384 * sizeof(float), stream);
    scatter_kernel<384, false><<<(unsigned)(((long)E * 96 + 255) / 256), blk, 0, stream>>>(
        hB, nullptr, src, dst, agg, E);
    gemm_kernel<0><<<dim3(256 / 16, gy), blk, 0, stream>>>(agg, hB, rden, sw1, sb1, hA, 384, 256, 1);

    // ---- stage 1: GraphConv 256 -> 256 ----
    hipMemsetAsync(agg, 0, (size_t)N * 256 * sizeof(float), stream);
    scatter_kernel<256, true><<<(unsigned)(((long)E * 64 + 255) / 256), blk, 0, stream>>>(
        hA, inv_src, src, dst, agg, E);
    gemm_kernel<1><<<dim3(256 / 16, gy), blk, 0, stream>>>(agg, nullptr, inv_dst, gw1, gb1, hB, 256, 256, 1);

    // ---- stage 2: SAGE 256 -> 128 ----
    hipMemsetAsync(agg, 0, (size_t)N * 256 * sizeof(float), stream);
    scatter_kernel<256, false><<<(unsigned)(((long)E * 64 + 255) / 256), blk, 0, stream>>>(
        hB, nullptr, src, dst, agg, E);
    gemm_kernel<0><<<dim3(128 / 16, gy), blk, 0, stream>>>(agg, hB, rden, sw2, sb2, hA, 256, 128, 1);

    // ---- stage 2: GraphConv 128 -> 128 ----
    hipMemsetAsync(agg, 0, (size_t)N * 128 * sizeof(float), stream);
    scatter_kernel<128, true><<<(unsigned)(((long)E * 32 + 255) / 256), blk, 0, stream>>>(
        hA, inv_src, src, dst, agg, E);
    gemm_kernel<1><<<dim3(128 / 16, gy), blk, 0, stream>>>(agg, nullptr, inv_dst, gw2, gb2, hB, 128, 128, 1);

    // ---- final FC 128 -> 128 (no ReLU) ----
    gemm_kernel<2><<<dim3(128 / 16, gy), blk, 0, stream>>>(hB, nullptr, nullptr, fcw, fcb,
                                                           (float*)d_out, 128, 128, 0);
}